// FlatSheafLearner_39436389712333
// MI455X (gfx1250) — compile-verified
//
#include <hip/hip_runtime.h>
#include <hip/hip_bf16.h>
#include <math.h>

// ---------------------------------------------------------------------------
// FlatSheafLearner forward on MI455X (gfx1250, wave32).
//   h  = gelu(x @ W1 + b1)                         (100000x128)(128x64)
//   L0: agg = scatter_add(h[src] -> dst)
//       h = gelu(h@Ws0 + agg@Wn0) + h  ==  gelu([h|agg] @ [Ws0;Wn0]) + h
//   L1: same with Ws1/Wn1
//   out = tanh(h @ W2 + b2)                        (100000x64)(64x2)
// GEMMs run on V_WMMA_F32_16X16X4_F32 (f32 end-to-end; the workload is
// memory-bound on the edge scatter, ~1.6-2 GB of HBM traffic per call).
// GEMM kernel is templated on the K-split so A-operand addressing is pure
// immediate-offset global_load_b64 (no per-iteration cndmask/compare).
// ---------------------------------------------------------------------------

typedef __attribute__((ext_vector_type(2))) float v2f;
typedef __attribute__((ext_vector_type(8))) float v8f;

#define HD  64            // hidden width
#define KT  128           // total K of the fused GEMMs (always 128 here)
#define LDSP 72           // padded LDS row stride (floats): halves of a wave
                          // hit disjoint bank sets (8k+n vs 8k+n+16)

__device__ __forceinline__ float gelu_exact(float v) {
    return 0.5f * v * (1.0f + erff(v * 0.70710678118654752f));
}

// C(M x 64) = gelu( [A0 | A1](M x 128) @ [W0 ; Wr](128 x 64) + bias ) + resid
//   rows k in [0,K0)   of W come from W0, A element from A0 (stride LDA0)
//   rows k in [K0,128) of W come from Wr, A element from A1 (stride LDA1)
// Block = 128 threads (4 waves), each wave owns one 16x16 tile of C;
// block covers rows [16*bid, 16*bid+16) x all 64 cols.
template <int K0, int LDA0, int LDA1, bool HAS_BIAS, bool HAS_RES>
__global__ void __launch_bounds__(128)
fused_wmma_gemm(const float* __restrict__ A0,
                const float* __restrict__ A1,
                const float* __restrict__ W0, const float* __restrict__ Wr,
                const float* __restrict__ bias,
                const float* __restrict__ resid,
                float* __restrict__ out, int M) {
    __shared__ float lds[KT * LDSP];      // 36 KB, 128x64 W tile, padded rows

    // ---- stage W into LDS: 2048 float4, 16 per thread --------------------
    const int tid = threadIdx.x;
    #pragma unroll
    for (int j = 0; j < 16; ++j) {
        int idx = tid + j * 128;          // float4 id 0..2047
        int k   = idx >> 4;               // W row 0..127
        int n4  = idx & 15;               // float4 within row
        const float* s;
        if (k < K0) s = W0 + (size_t)k * HD + n4 * 4;
        else        s = Wr + (size_t)(k - K0) * HD + n4 * 4;
        float4 v = *(const float4*)s;
        *(float4*)&lds[k * LDSP + n4 * 4] = v;
    }
    __syncthreads();

    const int lane = tid & 31;
    const int half = lane >> 4;           // 0: lanes 0-15, 1: lanes 16-31
    const int l16  = lane & 15;
    const int wave = tid >> 5;            // tile_n = wave (cols wave*16..)
    const int n    = wave * 16 + l16;     // global column 0..63
    int m = blockIdx.x * 16 + l16;        // A row for this lane
    if (m >= M) m = M - 1;                // clamp (keeps EXEC all-1s)

    // per-lane A base pointers (all per-iteration addressing is immediate)
    const float* a0p = A0 + (size_t)m * LDA0 + half * 2;
    const float* a1p = nullptr;
    if constexpr (K0 < KT) a1p = A1 + (size_t)m * LDA1 + half * 2;

    v8f c = {};
    #pragma unroll
    for (int kk = 0; kk < KT; kk += 4) {
        const int k0 = kk + half * 2;     // this lane's K pair {k0, k0+1}
        v2f a, b;
        if constexpr (true) {
            if (kk < K0) {                // folds: K0 is a template constant
                a.x = a0p[kk];
                a.y = a0p[kk + 1];
            } else {
                a.x = a1p[kk - K0];
                a.y = a1p[kk - K0 + 1];
            }
        }
        b.x = lds[k0 * LDSP + n];
        b.y = lds[(k0 + 1) * LDSP + n];
        // D = A(16x4) * B(4x16) + C   -> v_wmma_f32_16x16x4_f32
        c = __builtin_amdgcn_wmma_f32_16x16x4_f32(
                false, a, false, b, (short)0, c, false, false);
    }

    // all K-loop reads of h rows done before any in-place epilogue write
    __syncthreads();

    const int rowbase = blockIdx.x * 16 + half * 8;
    #pragma unroll
    for (int i = 0; i < 8; ++i) {
        int row = rowbase + i;            // C layout: vgpr i -> M = i + 8*half
        if (row < M) {
            float v = c[i];
            if constexpr (HAS_BIAS) v += bias[n];
            v = gelu_exact(v);
            size_t o = (size_t)row * HD + n;
            if constexpr (HAS_RES) v += resid[o];
            out[o] = v;
        }
    }
}

// agg[dst[e]] += h[src[e]] over 64 features; 16 threads/edge, float4 gather,
// 4x global_atomic_add_f32 scatter.
__global__ void __launch_bounds__(256)
edge_scatter_add(const float* __restrict__ h,
                 const int* __restrict__ src, const int* __restrict__ dst,
                 float* __restrict__ agg, int E) {
    int t = blockIdx.x * blockDim.x + threadIdx.x;
    int e = t >> 4;
    if (e >= E) return;
    int f = (t & 15) << 2;                // feature chunk 0,4,...,60
    int s = src[e];
    int d = dst[e];
    float4 v = *(const float4*)(h + (size_t)s * HD + f);
    float* p = agg + (size_t)d * HD + f;
    atomicAdd(p + 0, v.x);
    atomicAdd(p + 1, v.y);
    atomicAdd(p + 2, v.z);
    atomicAdd(p + 3, v.w);
}

// out(M x 2) = tanh(h @ W2 + b2)
__global__ void __launch_bounds__(256)
head_tanh(const float* __restrict__ h, const float* __restrict__ W2,
          const float* __restrict__ b2, float* __restrict__ out, int M) {
    int r = blockIdx.x * blockDim.x + threadIdx.x;
    if (r >= M) return;
    const float* hr = h + (size_t)r * HD;
    float a0 = 0.f, a1 = 0.f;
    #pragma unroll
    for (int k = 0; k < HD; k += 4) {
        float4 v = *(const float4*)(hr + k);
        a0 = fmaf(v.x, W2[(k + 0) * 2 + 0], a0);
        a1 = fmaf(v.x, W2[(k + 0) * 2 + 1], a1);
        a0 = fmaf(v.y, W2[(k + 1) * 2 + 0], a0);
        a1 = fmaf(v.y, W2[(k + 1) * 2 + 1], a1);
        a0 = fmaf(v.z, W2[(k + 2) * 2 + 0], a0);
        a1 = fmaf(v.z, W2[(k + 2) * 2 + 1], a1);
        a0 = fmaf(v.w, W2[(k + 3) * 2 + 0], a0);
        a1 = fmaf(v.w, W2[(k + 3) * 2 + 1], a1);
    }
    out[(size_t)r * 2 + 0] = tanhf(a0 + b2[0]);
    out[(size_t)r * 2 + 1] = tanhf(a1 + b2[1]);
}

extern "C" void kernel_launch(void* const* d_in, const int* in_sizes, int n_in,
                              void* d_out, int out_size, void* d_ws, size_t ws_size,
                              hipStream_t stream) {
    const float* x   = (const float*)d_in[0];
    const int*   ei  = (const int*)d_in[1];   // [2, E]: src row then dst row
    const float* W1  = (const float*)d_in[2];
    const float* b1  = (const float*)d_in[3];
    const float* Ws0 = (const float*)d_in[4];
    const float* Wn0 = (const float*)d_in[5];
    const float* Ws1 = (const float*)d_in[6];
    const float* Wn1 = (const float*)d_in[7];
    const float* W2  = (const float*)d_in[8];
    const float* b2  = (const float*)d_in[9];
    float* out = (float*)d_out;

    const int M = in_sizes[0] / 128;          // 100000 nodes
    const int E = in_sizes[1] / 2;            // 1600000 edges
    const int* src = ei;
    const int* dst = ei + E;

    float* h   = (float*)d_ws;                // M*64 floats
    float* agg = h + (size_t)M * HD;          // M*64 floats
    const size_t aggBytes = (size_t)M * HD * sizeof(float);

    const int mtiles = (M + 15) / 16;         // 6250 (exact)
    const int sblk   = (E * 16 + 255) / 256;  // scatter grid

    // emb1: h = gelu(x @ W1 + b1)
    fused_wmma_gemm<128, 128, 0, true, false><<<mtiles, 128, 0, stream>>>(
        x, nullptr, W1, W1, b1, nullptr, h, M);

    // layer 0
    hipMemsetAsync(agg, 0, aggBytes, stream);
    edge_scatter_add<<<sblk, 256, 0, stream>>>(h, src, dst, agg, E);
    fused_wmma_gemm<64, 64, 64, false, true><<<mtiles, 128, 0, stream>>>(
        h, agg, Ws0, Wn0, nullptr, h, h, M);

    // layer 1
    hipMemsetAsync(agg, 0, aggBytes, stream);
    edge_scatter_add<<<sblk, 256, 0, stream>>>(h, src, dst, agg, E);
    fused_wmma_gemm<64, 64, 64, false, true><<<mtiles, 128, 0, stream>>>(
        h, agg, Ws1, Wn1, nullptr, h, h, M);

    // head: out = tanh(h @ W2 + b2)
    head_tanh<<<(M + 255) / 256, 256, 0, stream>>>(h, W2, b2, out, M);
}